// DeepGAT_83193516524093
// MI455X (gfx1250) — compile-verified
//
#include <hip/hip_runtime.h>
#include <math.h>

// ---------------- problem constants (match reference) ----------------
#define NNODES 100000
#define NEDGES 1600000
#define ETOT   (NEDGES + NNODES)   // edges + self loops
#define INFEAT 256
#define HIDF   128
#define NHEADS 8
#define CHEAD  16
#define NLAYER 3
#define BNEPS  1e-5f
#define SLOPE  0.2f
#define LDSPAD 4                   // floats; keeps 16B alignment + bank spread

typedef float v2f __attribute__((ext_vector_type(2)));
typedef float v8f __attribute__((ext_vector_type(8)));

__device__ __forceinline__ float elu_f(float x)   { return x > 0.f ? x : expm1f(x); }
__device__ __forceinline__ float lrelu_f(float x) { return x > 0.f ? x : SLOPE * x; }

// ---------------------------------------------------------------------
// WMMA f32 GEMM: C[nrows x 128] = A[nrows x K] * B[K x 128] (+bias, ELU)
// Block: 8 waves (32,8). Each block = 16 output rows, each wave = 16 cols.
// A strip staged straight into LDS with GLOBAL_LOAD_ASYNC_TO_LDS_B128
// (ASYNCcnt-tracked CDNA5 path). Row stride K+4 floats:
//   - 16B-aligned rows (async b128 requirement)
//   - (K+4) % 64 == 4 for K in {128,256} -> WMMA column reads hit banks
//     (4*row + k [+2]) mod 64: all 32 lanes distinct, conflict-free.
//
// V_WMMA_F32_16X16X4_F32 operand layouts (ISA 7.12.2):
//   A 16x4:  lanes 0-15  -> row=lane,    v[0]=K0, v[1]=K1
//            lanes 16-31 -> row=lane-16, v[0]=K2, v[1]=K3
//   B 4x16:  mirrored: lane%16 = col, half selects K pair
//   C/D 16x16: vgpr i: lanes 0-15 -> M=i, N=lane; lanes 16-31 -> M=8+i
// ---------------------------------------------------------------------
template<int K, bool BIAS_ELU>
__global__ __launch_bounds__(256)
void wmma_gemm_f32(const float* __restrict__ A, const float* __restrict__ B,
                   const float* __restrict__ bias, float* __restrict__ C,
                   int nrows)
{
    __shared__ float lds[16 * (K + LDSPAD)];
    const int r0   = blockIdx.x * 16;
    const int lane = threadIdx.x;
    const int wave = threadIdx.y;
    const int tid  = wave * 32 + lane;

    // LDS byte offset of the staging buffer (addrspace(3) offset = low 32 bits)
    const unsigned ldsbase = (unsigned)(uintptr_t)&lds[0];

    // async-stage A tile [16 x K] into LDS, one b128 chunk per iteration
    for (int i = tid * 4; i < 16 * K; i += 256 * 4) {
        const int row = i / K;
        const int col = i - row * K;
        int gr = r0 + row;
        if (gr >= nrows) gr = nrows - 1;    // N % 16 == 0 -> never taken
        const float*   gp     = &A[(size_t)gr * K + col];
        const unsigned ldsoff = ldsbase + (unsigned)(row * (K + LDSPAD) + col) * 4u;
        asm volatile("global_load_async_to_lds_b128 %0, %1, off"
                     :: "v"(ldsoff), "v"(gp) : "memory");
    }
    asm volatile("s_wait_asynccnt 0x0" ::: "memory");
    __syncthreads();

    const int c0   = wave * 16;
    const int half = lane >> 4;     // 0: K pair {0,1}, 1: K pair {2,3}
    const int lrow = lane & 15;

    v8f acc = {0.f, 0.f, 0.f, 0.f, 0.f, 0.f, 0.f, 0.f};
    #pragma unroll 4
    for (int k = 0; k < K; k += 4) {
        v2f a, b;
        const float* ar = &lds[lrow * (K + LDSPAD) + k + 2 * half];
        a.x = ar[0];
        a.y = ar[1];
        const float* br = &B[(size_t)(k + 2 * half) * HIDF + c0 + lrow];
        b.x = br[0];
        b.y = br[HIDF];
        acc = __builtin_amdgcn_wmma_f32_16x16x4_f32(
            /*neg_a=*/false, a, /*neg_b=*/false, b,
            /*c_mod=*/(short)0, acc, /*reuse_a=*/false, /*reuse_b=*/false);
    }

    #pragma unroll
    for (int i = 0; i < 8; ++i) {
        const int gr = r0 + i + 8 * half;
        const int gc = c0 + lrow;
        if (gr < nrows) {
            float v = acc[i];
            if (BIAS_ELU) v = elu_f(v + bias[gc]);
            C[(size_t)gr * HIDF + gc] = v;
        }
    }
}

// ---------------- attention logits: al = einsum(nhc,hc->nh) ----------------
__global__ __launch_bounds__(256)
void att_logits_kernel(const float* __restrict__ hW,
                       const float* __restrict__ a_src,
                       const float* __restrict__ a_dst,
                       float* __restrict__ als, float* __restrict__ ald)
{
    const int i = blockIdx.x * blockDim.x + threadIdx.x;   // node*NHEADS + head
    if (i >= NNODES * NHEADS) return;
    const int n  = i / NHEADS;
    const int hd = i - n * NHEADS;
    const float4* hv = (const float4*)&hW[(size_t)n * HIDF + hd * CHEAD];
    const float4* as = (const float4*)&a_src[hd * CHEAD];
    const float4* ad = (const float4*)&a_dst[hd * CHEAD];
    float s = 0.f, d = 0.f;
    #pragma unroll
    for (int j = 0; j < 4; ++j) {
        const float4 h4 = hv[j], a4 = as[j], b4 = ad[j];
        s += h4.x * a4.x + h4.y * a4.y + h4.z * a4.z + h4.w * a4.w;
        d += h4.x * b4.x + h4.y * b4.y + h4.z * b4.z + h4.w * b4.w;
    }
    als[i] = s;
    ald[i] = d;
}

// ---------------- per-layer init: aggr=conv_b, m=-inf, denom=0 -------------
__global__ __launch_bounds__(256)
void init_aggr_kernel(const float* __restrict__ conv_b, float* __restrict__ aggr)
{
    const int i = blockIdx.x * blockDim.x + threadIdx.x;
    if (i < NNODES * HIDF) aggr[i] = conv_b[i & (HIDF - 1)];
}

__global__ __launch_bounds__(256)
void init_mdenom_kernel(float* __restrict__ m, float* __restrict__ den)
{
    const int i = blockIdx.x * blockDim.x + threadIdx.x;
    if (i < NNODES * NHEADS) { m[i] = -INFINITY; den[i] = 0.f; }
}

// ---------------- edge helpers ---------------------------------------------
__device__ __forceinline__ void edge_nodes(const int* __restrict__ ei, int e,
                                           int& s, int& d)
{
    if (e < NEDGES) { s = ei[e]; d = ei[NEDGES + e]; }
    else            { s = e - NEDGES; d = s; }        // self loop
}

// ---------------- edge pass 1: segment max of leaky-relu logits ------------
__global__ __launch_bounds__(256)
void edge_max_kernel(const int* __restrict__ ei, const float* __restrict__ als,
                     const float* __restrict__ ald, float* __restrict__ m)
{
    const int i = blockIdx.x * blockDim.x + threadIdx.x;
    if (i >= ETOT * NHEADS) return;
    const int e = i >> 3, hd = i & 7;
    if (e + 2048 < NEDGES) {                 // warm index stream ahead
        __builtin_prefetch(&ei[e + 2048], 0, 1);
        __builtin_prefetch(&ei[NEDGES + e + 2048], 0, 1);
    }
    int s, d; edge_nodes(ei, e, s, d);
    const float ev = lrelu_f(als[s * NHEADS + hd] + ald[d * NHEADS + hd]);
    atomicMax(&m[d * NHEADS + hd], ev);
}

// ---------------- edge pass 2: sum of exp(e - max) -------------------------
__global__ __launch_bounds__(256)
void edge_expsum_kernel(const int* __restrict__ ei, const float* __restrict__ als,
                        const float* __restrict__ ald, const float* __restrict__ m,
                        float* __restrict__ den)
{
    const int i = blockIdx.x * blockDim.x + threadIdx.x;
    if (i >= ETOT * NHEADS) return;
    const int e = i >> 3, hd = i & 7;
    if (e + 2048 < NEDGES) {
        __builtin_prefetch(&ei[e + 2048], 0, 1);
        __builtin_prefetch(&ei[NEDGES + e + 2048], 0, 1);
    }
    int s, d; edge_nodes(ei, e, s, d);
    const float ev = lrelu_f(als[s * NHEADS + hd] + ald[d * NHEADS + hd]);
    atomicAdd(&den[d * NHEADS + hd], __expf(ev - m[d * NHEADS + hd]));
}

// ---------------- edge pass 3: weighted message scatter --------------------
// one wave32 per edge; lane handles 4 channels of head lane/4
__global__ __launch_bounds__(256)
void edge_message_kernel(const int* __restrict__ ei, const float* __restrict__ hW,
                         const float* __restrict__ als, const float* __restrict__ ald,
                         const float* __restrict__ m, const float* __restrict__ den,
                         float* __restrict__ aggr)
{
    const int gid  = blockIdx.x * blockDim.x + threadIdx.x;
    const int e    = gid >> 5;
    if (e >= ETOT) return;
    const int lane = gid & 31;
    if (lane == 0 && e + 256 < NEDGES) {
        __builtin_prefetch(&ei[e + 256], 0, 1);
        __builtin_prefetch(&ei[NEDGES + e + 256], 0, 1);
    }
    int s, d; edge_nodes(ei, e, s, d);
    const int hd = lane >> 2;     // 4 lanes per head (16 channels)
    const float ev    = lrelu_f(als[s * NHEADS + hd] + ald[d * NHEADS + hd]);
    const float alpha = __expf(ev - m[d * NHEADS + hd]) /
                        (den[d * NHEADS + hd] + 1e-16f);
    const float4 hv = *(const float4*)&hW[(size_t)s * HIDF + lane * 4];
    float* op = &aggr[(size_t)d * HIDF + lane * 4];
    atomicAdd(op + 0, hv.x * alpha);
    atomicAdd(op + 1, hv.y * alpha);
    atomicAdd(op + 2, hv.z * alpha);
    atomicAdd(op + 3, hv.w * alpha);
}

// ---------------- batchnorm stats: per-channel sum / sumsq -----------------
#define BN_NODES_PER_BLK 256
__global__ __launch_bounds__(128)
void bn_stats_kernel(const float* __restrict__ aggr, float* __restrict__ sum,
                     float* __restrict__ sq)
{
    const int ch = threadIdx.x;                  // 0..127
    const int n0 = blockIdx.x * BN_NODES_PER_BLK;
    const int n1 = min(n0 + BN_NODES_PER_BLK, NNODES);
    float s = 0.f, q = 0.f;
    for (int n = n0; n < n1; ++n) {
        const float v = aggr[(size_t)n * HIDF + ch];
        s += v; q += v * v;
    }
    atomicAdd(&sum[ch], s);
    atomicAdd(&sq[ch],  q);
}

// ---------------- bn + elu + residual (in place into h) --------------------
__global__ __launch_bounds__(256)
void bn_apply_kernel(const float* __restrict__ aggr, const float* __restrict__ sum,
                     const float* __restrict__ sq, const float* __restrict__ g,
                     const float* __restrict__ b, float* __restrict__ h)
{
    const int i = blockIdx.x * blockDim.x + threadIdx.x;
    if (i >= NNODES * HIDF) return;
    const int ch   = i & (HIDF - 1);
    const float mu = sum[ch] * (1.f / NNODES);
    const float vr = sq[ch] * (1.f / NNODES) - mu * mu;
    const float v  = (aggr[i] - mu) * rsqrtf(vr + BNEPS) * g[ch] + b[ch];
    h[i] = elu_f(v) + h[i];
}

// ---------------- final fc: [N,128] @ [128,1] + b --------------------------
__global__ __launch_bounds__(256)
void fc_kernel(const float* __restrict__ h, const float* __restrict__ w,
               const float* __restrict__ b, float* __restrict__ out)
{
    const int n = blockIdx.x * blockDim.x + threadIdx.x;
    if (n >= NNODES) return;
    const float4* hv = (const float4*)&h[(size_t)n * HIDF];
    const float4* wv = (const float4*)w;
    float acc = 0.f;
    #pragma unroll
    for (int j = 0; j < HIDF / 4; ++j) {
        const float4 a = hv[j], c = wv[j];
        acc += a.x * c.x + a.y * c.y + a.z * c.z + a.w * c.w;
    }
    out[n] = acc + b[0];
}

// ---------------------------------------------------------------------------
extern "C" void kernel_launch(void* const* d_in, const int* in_sizes, int n_in,
                              void* d_out, int out_size, void* d_ws, size_t ws_size,
                              hipStream_t stream)
{
    (void)in_sizes; (void)n_in; (void)out_size; (void)ws_size;
    const float* x       = (const float*)d_in[0];
    const int*   ei      = (const int*)  d_in[1];
    const float* proj_w  = (const float*)d_in[2];
    const float* proj_b  = (const float*)d_in[3];
    const float* W       = (const float*)d_in[4];   // [L,128,128]
    const float* att_src = (const float*)d_in[5];   // [L,8,16]
    const float* att_dst = (const float*)d_in[6];
    const float* conv_b  = (const float*)d_in[7];   // [L,128]
    const float* bn_g    = (const float*)d_in[8];
    const float* bn_b    = (const float*)d_in[9];
    const float* fc_w    = (const float*)d_in[10];  // [128]
    const float* fc_b    = (const float*)d_in[11];
    float*       out     = (float*)d_out;

    // workspace carve-up (f32)
    float* h     = (float*)d_ws;                         // N*128
    float* hW    = h     + (size_t)NNODES * HIDF;        // N*128
    float* aggr  = hW    + (size_t)NNODES * HIDF;        // N*128
    float* als   = aggr  + (size_t)NNODES * HIDF;        // N*8
    float* ald   = als   + (size_t)NNODES * NHEADS;      // N*8
    float* mmax  = ald   + (size_t)NNODES * NHEADS;      // N*8
    float* den   = mmax  + (size_t)NNODES * NHEADS;      // N*8
    float* bnsum = den   + (size_t)NNODES * NHEADS;      // 2*128 (sum|sumsq)

    const dim3 gemm_blk(32, 8);
    const int  row_tiles = (NNODES + 15) / 16;

    // input projection + ELU
    wmma_gemm_f32<INFEAT, true><<<row_tiles, gemm_blk, 0, stream>>>(
        x, proj_w, proj_b, h, NNODES);

    for (int l = 0; l < NLAYER; ++l) {
        // h @ W[l]
        wmma_gemm_f32<HIDF, false><<<row_tiles, gemm_blk, 0, stream>>>(
            h, W + (size_t)l * HIDF * HIDF, nullptr, hW, NNODES);

        att_logits_kernel<<<(NNODES * NHEADS + 255) / 256, 256, 0, stream>>>(
            hW, att_src + l * NHEADS * CHEAD, att_dst + l * NHEADS * CHEAD, als, ald);

        init_aggr_kernel<<<(NNODES * HIDF + 255) / 256, 256, 0, stream>>>(
            conv_b + l * HIDF, aggr);
        init_mdenom_kernel<<<(NNODES * NHEADS + 255) / 256, 256, 0, stream>>>(
            mmax, den);

        edge_max_kernel<<<((size_t)ETOT * NHEADS + 255) / 256, 256, 0, stream>>>(
            ei, als, ald, mmax);
        edge_expsum_kernel<<<((size_t)ETOT * NHEADS + 255) / 256, 256, 0, stream>>>(
            ei, als, ald, mmax, den);
        edge_message_kernel<<<((size_t)ETOT * 32 + 255) / 256, 256, 0, stream>>>(
            ei, hW, als, ald, mmax, den, aggr);

        hipMemsetAsync(bnsum, 0, 2 * HIDF * sizeof(float), stream);
        bn_stats_kernel<<<(NNODES + BN_NODES_PER_BLK - 1) / BN_NODES_PER_BLK,
                          128, 0, stream>>>(aggr, bnsum, bnsum + HIDF);
        bn_apply_kernel<<<(NNODES * HIDF + 255) / 256, 256, 0, stream>>>(
            aggr, bnsum, bnsum + HIDF, bn_g + l * HIDF, bn_b + l * HIDF, h);
    }

    fc_kernel<<<(NNODES + 255) / 256, 256, 0, stream>>>(h, fc_w, fc_b, out);
}